// UncertaintyDynamicQAgent_33681133535603
// MI455X (gfx1250) — compile-verified
//
#include <hip/hip_runtime.h>
#include <stdint.h>

#define GLOBAL_AS __attribute__((address_space(1)))
#define LDS_AS    __attribute__((address_space(3)))

typedef int v4i_t __attribute__((vector_size(4 * sizeof(int))));

#define NSESS_PER_BLOCK 256
#define TTILE   8
#define NTRIALS 1024
#define NTILES  (NTRIALS / TTILE)   // 128 tiles of 8 trials
#define LSTRIDE 28                  // floats per session strip in LDS (112 B, 16B aligned)

// ---------------- CDNA5 async-copy + split-counter helpers ----------------

__device__ __forceinline__ void async_load_b128(const float* gsrc, float* ldst) {
#if __has_builtin(__builtin_amdgcn_global_load_async_to_lds_b128)
  __builtin_amdgcn_global_load_async_to_lds_b128(
      (GLOBAL_AS v4i_t*)gsrc, (LDS_AS v4i_t*)ldst, /*offset=*/0, /*cpol=*/0);
#else
  unsigned lds_off = (unsigned)(unsigned long long)(LDS_AS char*)ldst;
  asm volatile("global_load_async_to_lds_b128 %0, %1, off"
               :: "v"(lds_off), "v"(gsrc) : "memory");
#endif
}

template <int N>
__device__ __forceinline__ void wait_asynccnt() {
#if __has_builtin(__builtin_amdgcn_s_wait_asynccnt)
  __builtin_amdgcn_s_wait_asynccnt(N);
#else
  asm volatile("s_wait_asynccnt %0" :: "n"(N) : "memory");
#endif
  asm volatile("" ::: "memory");  // compiler reorder fence
}

__device__ __forceinline__ void wait_dscnt0() {
#if __has_builtin(__builtin_amdgcn_s_wait_dscnt)
  __builtin_amdgcn_s_wait_dscnt(0);
#else
  asm volatile("s_wait_dscnt 0" ::: "memory");
#endif
  asm volatile("" ::: "memory");
}

// ---------------- model math ----------------

// smooth_clamp(x, 0, 1) with beta = 100, JAX-stable softplus:
//   sp(z) = (max(y,0) + log1p(exp(-|y|))) / 100,  y = 100 z
// Uses hardware v_exp_f32 / v_log_f32 (__expf/__logf): with e = exp(-|y|) in
// (0,1], log(1+e) vs log1p(e) differs by <= ~6e-8 abs, scaled by 0.01 ->
// ~1e-9 absolute error in the clamp output. Keeps 3 transcendental chains
// per step short instead of ~30-instr libm polynomials on the critical path.
__device__ __forceinline__ float sclamp01(float x) {
  float z  = (x < 0.5f) ? x : (1.0f - x);           // sp argument
  float y  = 100.0f * z;
  float e  = __expf(-fabsf(y));                     // v_exp_f32 path
  float sp = (fmaxf(y, 0.0f) + __logf(1.0f + e)) * 0.01f;  // v_log_f32 path
  return (x < 0.5f) ? sp : (1.0f - sp);
}

__global__ __launch_bounds__(NSESS_PER_BLOCK)
void qagent_scan_kernel(const float* __restrict__ in,
                        const float* __restrict__ alpha0s,
                        const float* __restrict__ gamma_alphas,
                        const float* __restrict__ gamma_lams,
                        const float* __restrict__ k_vals,
                        float* __restrict__ out,
                        int n_sess) {
  __shared__ __align__(16) float smem[2][NSESS_PER_BLOCK * LSTRIDE];

  const int tid = threadIdx.x;
  int s = blockIdx.x * NSESS_PER_BLOCK + tid;
  const bool active = (s < n_sess);
  if (!active) s = n_sess - 1;   // clamp: keep wave uniform, predicate stores only

  // Uniform parameters -> scalar loads
  const float a0 = alpha0s[0], a1 = alpha0s[1], a2 = alpha0s[2], a3 = alpha0s[3];
  const float ga0 = gamma_alphas[0], ga1 = gamma_alphas[1],
              ga2 = gamma_alphas[2], ga3 = gamma_alphas[3];
  const float gl0 = gamma_lams[0], gl1 = gamma_lams[1],
              gl2 = gamma_lams[2], gl3 = gamma_lams[3];
  const float k0 = k_vals[0], k1 = k_vals[1], k2 = k_vals[2], k3 = k_vals[3];

  const float* gbase = in  + (size_t)s * (NTRIALS * 3);
  float*       obase = out + (size_t)s * (NTRIALS * 2);
  float* lb0 = &smem[0][tid * LSTRIDE];
  float* lb1 = &smem[1][tid * LSTRIDE];

  // Prime the pipeline: async-prefetch tiles 0 and 1 (ASYNCcnt += 12)
#pragma unroll
  for (int j = 0; j < 6; ++j) async_load_b128(gbase + 0 * 24 + j * 4, lb0 + j * 4);
#pragma unroll
  for (int j = 0; j < 6; ++j) async_load_b128(gbase + 1 * 24 + j * 4, lb1 + j * 4);

  // carry state
  float Q0 = 0.0f, Q1 = 0.0f, lam0 = 0.5f, lam1 = 0.5f, alpha = 0.0f;

  for (int kt = 0; kt < NTILES; ++kt) {
    // tile kt complete: allow only the next tile's 6 loads outstanding
    if (kt == NTILES - 1) wait_asynccnt<0>(); else wait_asynccnt<6>();

    const float* lb = (kt & 1) ? lb1 : lb0;
    float xt[24];
#pragma unroll
    for (int j = 0; j < 6; ++j) {
      float4 v = *reinterpret_cast<const float4*>(lb + j * 4);  // ds_load_b128
      xt[4 * j + 0] = v.x; xt[4 * j + 1] = v.y;
      xt[4 * j + 2] = v.z; xt[4 * j + 3] = v.w;
    }
    wait_dscnt0();  // LDS strip drained into VGPRs -> safe to overwrite

    if (kt + 2 < NTILES) {  // async-prefetch tile kt+2 into the other buffer, not the one just read
      const float* gsrc = gbase + (kt + 2) * 24;
      float* ld = (kt & 1) ? lb0 : lb1;
#pragma unroll
      for (int j = 0; j < 6; ++j) async_load_b128(gsrc + j * 4, ld + j * 4);
    }

    // 8 recurrence steps out of registers
#pragma unroll
    for (int i = 0; i < TTILE; ++i) {
      const float cl = xt[i * 3 + 0];
      const float cr = xt[i * 3 + 1];
      const float o  = xt[i * 3 + 2];

      const float sL0 = cl * o, sL1 = cl * (1.0f - o);
      const float sL2 = cr * o, sL3 = cr * (1.0f - o);
      // sel_R = sel_L[perm], perm = [2,3,0,1]
      const float sR0 = sL2, sR1 = sL3, sR2 = sL0, sR3 = sL1;

      const float dL0 = k0 - Q0, dL1 = k1 - Q0, dL2 = k2 - Q0, dL3 = k3 - Q0;
      const float dR0 = k0 - Q1, dR1 = k1 - Q1, dR2 = k2 - Q1, dR3 = k3 - Q1;

      const float lL0 = fabsf(dL0) - lam0, lL1 = fabsf(dL1) - lam0;
      const float lL2 = fabsf(dL2) - lam0, lL3 = fabsf(dL3) - lam0;
      const float lR0 = fabsf(dR0) - lam1, lR1 = fabsf(dR1) - lam1;
      const float lR2 = fabsf(dR2) - lam1, lR3 = fabsf(dR3) - lam1;

      const float ad0 = (a0 + lL0 - alpha) * sL0;
      const float ad1 = (a1 + lL1 - alpha) * sL1;
      const float ad2 = (a0 + lR0 - alpha) * sR2;
      const float ad3 = (a1 + lR1 - alpha) * sR3;
      const float alpha_delta = ad0 * ga0 + ad1 * ga1 + ad2 * ga2 + ad3 * ga3;
      const float alpha_first = a0 * sL0 + a1 * sL1 + a2 * sL2 + a3 * sL3;

      const bool first = (kt == 0) && (i == 0);
      const float alpha_new =
          first ? alpha_first : sclamp01(alpha + alpha_delta);

      const float lamLs = lam0 + gl0 * lL0 * sL0 + gl1 * lL1 * sL1
                               + gl2 * lL2 * sL2 + gl3 * lL3 * sL3;
      const float lamRs = lam1 + gl0 * lR0 * sR0 + gl1 * lR1 * sR1
                               + gl2 * lR2 * sR2 + gl3 * lR3 * sR3;
      const float lam0n = sclamp01(lamLs);
      const float lam1n = sclamp01(lamRs);

      const float sdL = dL0 * sL0 + dL1 * sL1 + dL2 * sL2 + dL3 * sL3;
      const float sdR = dR0 * sR0 + dR1 * sR1 + dR2 * sR2 + dR3 * sR3;

      const float qL = Q0 + alpha_new * (1.0f - lam0) * sdL;
      const float qR = Q1 + alpha_new * (1.0f - lam1) * sdR;

      if (active) {
        const int t = kt * TTILE + i;
        *reinterpret_cast<float2*>(obase + 2 * t) = make_float2(qL, qR);
      }

      Q0 = qL; Q1 = qR; lam0 = lam0n; lam1 = lam1n; alpha = alpha_new;
    }
  }
}

extern "C" void kernel_launch(void* const* d_in, const int* in_sizes, int n_in,
                              void* d_out, int out_size, void* d_ws, size_t ws_size,
                              hipStream_t stream) {
  const float* in  = (const float*)d_in[0];
  const float* a0s = (const float*)d_in[1];
  const float* gas = (const float*)d_in[2];
  const float* gls = (const float*)d_in[3];
  const float* kvs = (const float*)d_in[4];
  float* out = (float*)d_out;

  const int n_sess = in_sizes[0] / (NTRIALS * 3);  // 8192
  const int blocks = (n_sess + NSESS_PER_BLOCK - 1) / NSESS_PER_BLOCK;

  qagent_scan_kernel<<<blocks, NSESS_PER_BLOCK, 0, stream>>>(
      in, a0s, gas, gls, kvs, out, n_sess);
}